// Ricker_Predation_42210938585242
// MI455X (gfx1250) — compile-verified
//
#include <hip/hip_runtime.h>
#include <hip/hip_bf16.h>

// ---------------------------------------------------------------------------
// Ricker predator-prey recurrence, T = ~1M steps.
//   Pass 1 (parallel, WMMA f32 16x16x4): g1[i] = a1*(1 + bx1*t + cx1*t^2),
//                                        g2[i] = a2*(1 + bx2*t + cx2*t^2)
//     computed as D = A(2x4 coeffs) x B(4x16 = [1, t_j, t_j^2, 0] columns),
//     16 timesteps per WMMA, stored as float2 pairs in d_ws.
//   Pass 2 (inherently serial, 1 lane, fp64): the nonlinear recurrence with
//     two independent fp64 exps per step (ILP-overlapped by the compiler).
// ---------------------------------------------------------------------------

typedef float v2f __attribute__((ext_vector_type(2)));
typedef float v8f __attribute__((ext_vector_type(8)));

#ifndef __has_builtin
#define __has_builtin(x) 0
#endif

#if defined(__gfx1250__) && __has_builtin(__builtin_amdgcn_wmma_f32_16x16x4_f32)
#define RICKER_HAVE_WMMA_F32X4 1
#else
#define RICKER_HAVE_WMMA_F32X4 0
#endif

// --------------------------- Pass 1: forcing table --------------------------
// A-matrix (16x4, f32): lane L holds row M = L&15; lanes 0-15 carry K=0,1 in
// VGPR0,1 and lanes 16-31 carry K=2,3 (ISA 7.12.2, 32-bit A 16x4 layout).
//   row 0 = [a1, a1*bx1, a1*cx1, 0]   -> g1
//   row 1 = [a2, a2*bx2, a2*cx2, 0]   -> g2
//   rows 2..15 = 0
// B-matrix (4x16, f32): row striped across lanes within one VGPR:
//   VGPR0: lanes 0-15 = row K=0 (all ones), lanes 16-31 = row K=2 (t^2)
//   VGPR1: lanes 0-15 = row K=1 (t),       lanes 16-31 = row K=3 (zeros)
// D: VGPR0 lanes 0-15 = g1 for N=0..15, VGPR1 lanes 0-15 = g2.
__global__ void ricker_forcing_wmma(const float* __restrict__ Temp,
                                    const double* __restrict__ P,
                                    float2* __restrict__ G, int n) {
    const int lane   = threadIdx.x & 31;
    const int wave   = (int)((blockIdx.x * blockDim.x + threadIdx.x) >> 5);
    const int nwaves = (int)((gridDim.x * blockDim.x) >> 5);
    const int ntiles = (n + 15) >> 4;

    // Coefficients (fp64 host params folded to f32 once per wave).
    const float a1  = (float)P[0];
    const float a1b = (float)(P[0] * P[3]);
    const float a1c = (float)(P[0] * P[4]);
    const float a2  = (float)P[5];
    const float a2b = (float)(P[5] * P[8]);
    const float a2c = (float)(P[5] * P[9]);

    const int  laneM = lane & 15;
    const bool hi    = (lane >= 16);

    float A0 = 0.0f, A1v = 0.0f;
    if (laneM == 0) { A0 = hi ? a1c : a1;  A1v = hi ? 0.0f : a1b; }
    if (laneM == 1) { A0 = hi ? a2c : a2;  A1v = hi ? 0.0f : a2b; }

    for (int tile = wave; tile < ntiles; tile += nwaves) {
        const int   idx = (tile << 4) + laneM;
        const float t   = (idx < n) ? Temp[idx] : 0.0f;
        const float B0  = hi ? (t * t) : 1.0f;
        const float B1  = hi ? 0.0f    : t;

#if RICKER_HAVE_WMMA_F32X4
        v8f c = {};
        v2f av; av.x = A0; av.y = A1v;
        v2f bv; bv.x = B0; bv.y = B1;
        // 8 args: (neg_a, A, neg_b, B, c_mod, C, reuse_a, reuse_b)
        v8f d = __builtin_amdgcn_wmma_f32_16x16x4_f32(
            false, av, false, bv, (short)0, c, false, false);
        const float g1 = d[0];   // M=0 row, N=lane (lanes 0-15)
        const float g2 = d[1];   // M=1 row
#else
        // VALU fallback (same math) if the f32 x4 builtin is unavailable.
        const float g1 = a1 + a1b * t + a1c * t * t;
        const float g2 = a2 + a2b * t + a2c * t * t;
#endif
        if (!hi && idx < n) {
            G[idx] = make_float2(g1, g2);
        }
    }
}

// --------------------------- Pass 2: serial scan ----------------------------
__global__ void __launch_bounds__(32)
ricker_serial(const float2* __restrict__ G, const float* __restrict__ Temp,
              const double* __restrict__ P, float* __restrict__ out,
              int T, int useG) {
    if (threadIdx.x != 0) return;

    const double alpha1 = P[0], beta1 = P[1], gamma1 = P[2], bx1 = P[3], cx1 = P[4];
    const double alpha2 = P[5], beta2 = P[6], gamma2 = P[7], bx2 = P[8], cx2 = P[9];
    const double A1B1 = alpha1 * beta1, A1G1 = alpha1 * gamma1;
    const double A2B2 = alpha2 * beta2, A2G2 = alpha2 * gamma2;

    double n1 = 1.0, n2 = 1.0;
    out[0] = 1.0f;        // out[0, 0]
    out[T] = 1.0f;        // out[1, 0]

    const int n = T - 1;
    for (int i = 0; i < n; ++i) {
        double g1, g2;
        if (useG) {
            if (((i & 255) == 0) && (i + 2048 < n)) {
                // gfx1250 global_prefetch_b8: keep the g-stream ahead of the
                // fp64-exp dependence chain.
                __builtin_prefetch((const void*)(G + i + 2048), 0, 0);
            }
            const float2 gv = G[i];
            g1 = (double)gv.x;
            g2 = (double)gv.y;
        } else {
            const double t = (double)Temp[i];
            g1 = alpha1 * (1.0 + bx1 * t + cx1 * t * t);
            g2 = alpha2 * (1.0 + bx2 * t + cx2 * t * t);
        }
        // Synchronous update: both species read the previous state.
        const double e1 = exp(g1 - A1B1 * n1 - A1G1 * n2);
        const double e2 = exp(g2 - A2B2 * n2 - A2G2 * n1);
        n1 *= e1;
        n2 *= e2;
        out[i + 1]     = (float)n1;   // row 0
        out[T + i + 1] = (float)n2;   // row 1
    }
}

// ------------------------------- launcher -----------------------------------
extern "C" void kernel_launch(void* const* d_in, const int* in_sizes, int n_in,
                              void* d_out, int out_size, void* d_ws, size_t ws_size,
                              hipStream_t stream) {
    const float*  Temp = (const float*)d_in[0];
    const double* P    = (const double*)d_in[1];
    float*        out  = (float*)d_out;

    const int T = in_sizes[0];
    const int n = T - 1;

    float2*      G    = (float2*)d_ws;
    const size_t need = (size_t)(n > 0 ? n : 0) * sizeof(float2);
    const int useG    = (n > 0 && ws_size >= need) ? 1 : 0;

    if (useG) {
        const int threads       = 256;               // 8 wave32 per block
        const int wavesPerBlock = threads / 32;
        const int tiles         = (n + 15) / 16;
        int blocks = (tiles + wavesPerBlock - 1) / wavesPerBlock;
        if (blocks > 2048) blocks = 2048;
        if (blocks < 1)    blocks = 1;
        ricker_forcing_wmma<<<blocks, threads, 0, stream>>>(Temp, P, G, n);
    }
    ricker_serial<<<1, 32, 0, stream>>>(G, Temp, P, out, T, useG);
}